// AttnGCNLayer_1580547974048
// MI455X (gfx1250) — compile-verified
//
#include <hip/hip_runtime.h>
#include <hip/hip_bf16.h>
#include <stdint.h>

// AttnGCN layer for MI455X (gfx1250, wave32).
// Memory-bound (~430MB HBM traffic vs ~82 GFLOP), so full fp32 precision with
// V_WMMA_F32_16X16X4_F32 for every matmul, and GLOBAL_LOAD_ASYNC_TO_LDS_B128
// (ASYNCcnt-tracked memory->LDS DMA) for all tile staging.

#define B_C 8
#define N_C 2048
#define E_C 512
#define D_C 512

typedef __attribute__((ext_vector_type(2))) float v2f;
typedef __attribute__((ext_vector_type(8))) float v8f;

__device__ __forceinline__ v8f wmma4(v2f a, v2f b, v8f c) {
    // (neg_a, A, neg_b, B, c_mod, C, reuse_a, reuse_b)
    return __builtin_amdgcn_wmma_f32_16x16x4_f32(false, a, false, b, (short)0, c,
                                                 false, false);
}

// ---- gfx1250 async memory->LDS copy (16B per lane), tracked with ASYNCcnt ----
// Generic pointers into LDS carry the wave-relative LDS byte offset in their
// low 32 bits (flat shared aperture), which is what the VDST operand wants.
__device__ __forceinline__ uint32_t lds_byte_off(const void* p) {
    return (uint32_t)(uintptr_t)p;
}
__device__ __forceinline__ void async_copy_b128(uint32_t lds_off, const void* gptr) {
    asm volatile("global_load_async_to_lds_b128 %0, %1, off"
                 :: "v"(lds_off), "v"(gptr)
                 : "memory");
}
__device__ __forceinline__ void wait_async_all() {
    asm volatile("s_wait_asynccnt 0x0" ::: "memory");
}

constexpr int XS_STRIDE = 516;   // 516 % 64 == 4  -> conflict-free column reads; 516*4 % 16 == 0
constexpr int SS_STRIDE = 2068;  // 2068 % 64 == 20 -> row*20 distinct banks, m=0..15

// Stage a 16 x 512 fp32 tile (rows contiguous in global) into LDS with
// async B128 copies: 2048 16-byte chunks over 256 threads.
__device__ __forceinline__ void stage_tile_async(float* dst, const float* src,
                                                 int tid) {
    for (int i = tid; i < 2048; i += 256) {
        const int e = i << 2;           // element index within 16x512 tile
        const int r = e >> 9, c = e & 511;
        async_copy_b128(lds_byte_off(&dst[r * XS_STRIDE + c]),
                        &src[(size_t)r * E_C + c]);
    }
    wait_async_all();
    __syncthreads();
}

// ---------------------------------------------------------------------------
// Kernel 1: Q = X@Wq + bq ; K = X@Wk + bk      (X flattened to (B*N, E))
// One block = 16 output rows. 8 waves x 8 tiles cover 32 Q-tiles + 32 K-tiles.
// ---------------------------------------------------------------------------
__global__ void qk_proj_kernel(const float* __restrict__ X,
                               const float* __restrict__ Wq, const float* __restrict__ bq,
                               const float* __restrict__ Wk, const float* __restrict__ bk,
                               float* __restrict__ Q, float* __restrict__ K) {
    extern __shared__ float lds[];
    float* Xs = lds;  // 16 x XS_STRIDE
    const int rowbase = blockIdx.x * 16;
    const int tid = threadIdx.x;

    stage_tile_async(Xs, X + (size_t)rowbase * E_C, tid);

    const int lane = tid & 31, wave = tid >> 5;
    const int half = lane >> 4, l16 = lane & 15;

    for (int j = 0; j < 8; ++j) {
        const int id = wave * 8 + j;                  // uniform per wave
        const float* W; const float* bias; float* Out; int nt;
        if (id < 32) { W = Wq; bias = bq; Out = Q; nt = id; }
        else         { W = Wk; bias = bk; Out = K; nt = id - 32; }
        const int n0 = nt * 16 + l16;

        v8f acc = {};
        for (int k = 0; k < E_C; k += 4) {
            const int kb = k + 2 * half;
            v2f a = { Xs[l16 * XS_STRIDE + kb], Xs[l16 * XS_STRIDE + kb + 1] };
            v2f b = { W[(size_t)kb * D_C + n0], W[(size_t)(kb + 1) * D_C + n0] };
            acc = wmma4(a, b, acc);
        }
        const float bv = bias[n0];
#pragma unroll
        for (int v = 0; v < 8; ++v) {
            const int row = rowbase + v + 8 * half;
            Out[(size_t)row * D_C + n0] = acc[v] + bv;
        }
    }
}

// ---------------------------------------------------------------------------
// Kernel 2: fused masked attention for one (batch, 16-query-row) tile.
// Scores row-block (16 x 2048 fp32) lives entirely in the 320KB WGP LDS.
// ---------------------------------------------------------------------------
__global__ void attn_kernel(const float* __restrict__ Q, const float* __restrict__ K,
                            const float* __restrict__ V,   // node_fts
                            const float* __restrict__ rel, float* __restrict__ CTX) {
    extern __shared__ float lds[];
    float* Qs  = lds;                       // 16 x XS_STRIDE
    float* Ss  = Qs + 16 * XS_STRIDE;       // 16 x SS_STRIDE
    float* red = Ss + 16 * SS_STRIDE;       // 2 x 256

    const int b  = blockIdx.x >> 7;         // 128 row tiles per batch
    const int mt = blockIdx.x & 127;
    const int rowbase = b * N_C + mt * 16;  // flattened query row
    const int tid = threadIdx.x;

    stage_tile_async(Qs, Q + (size_t)rowbase * D_C, tid);

    const int lane = tid & 31, wave = tid >> 5;
    const int half = lane >> 4, l16 = lane & 15;

    // ---- Phase A: S = Q @ K^T (unscaled) ----
    for (int t = 0; t < 16; ++t) {
        const int kt = wave * 16 + t;                 // key tile 0..127
        const size_t keyrow = (size_t)(b * N_C + kt * 16 + l16) * D_C;
        v8f acc = {};
        for (int k = 0; k < D_C; k += 4) {
            const int kb = k + 2 * half;
            v2f a = { Qs[l16 * XS_STRIDE + kb], Qs[l16 * XS_STRIDE + kb + 1] };
            v2f bk = { K[keyrow + kb], K[keyrow + kb + 1] };
            acc = wmma4(a, bk, acc);
        }
#pragma unroll
        for (int v = 0; v < 8; ++v)
            Ss[(v + 8 * half) * SS_STRIDE + kt * 16 + l16] = acc[v];
    }
    __syncthreads();

    // ---- Phase B: masked softmax (16 threads per row, 128 cols each) ----
    {
        const int r = tid >> 4, seg = tid & 15;
        const float inv_sqrt_d = 0.044194173824159216f;  // 1/sqrt(512)
        const float* erow = rel + ((size_t)b * N_C + (size_t)(mt * 16 + r)) * N_C;

        float pmax = -3.0e38f;
        for (int c = seg * 128; c < seg * 128 + 128; ++c) {
            const float e = erow[c];
            const float s = (e != 0.0f) ? Ss[r * SS_STRIDE + c] * inv_sqrt_d : -1.0e30f;
            pmax = fmaxf(pmax, s);
        }
        red[r * 16 + seg] = pmax;
        __syncthreads();
        float m = -3.0e38f;
        for (int i = 0; i < 16; ++i) m = fmaxf(m, red[r * 16 + i]);

        float psum = 0.0f;
        for (int c = seg * 128; c < seg * 128 + 128; ++c) {
            const float e = erow[c];
            float p = 0.0f;
            if (e != 0.0f) p = expf(Ss[r * SS_STRIDE + c] * inv_sqrt_d - m);
            Ss[r * SS_STRIDE + c] = p;     // masked entries forced to 0
            psum += p;
        }
        red[256 + r * 16 + seg] = psum;
        __syncthreads();
        float tot = 0.0f;
        for (int i = 0; i < 16; ++i) tot += red[256 + r * 16 + i];
        const float inv = (tot > 0.0f) ? 1.0f / tot : 0.0f;
        for (int c = seg * 128; c < seg * 128 + 128; ++c)
            Ss[r * SS_STRIDE + c] *= inv;
    }
    __syncthreads();

    // ---- Phase C: CTX = attn @ V  (V loads coalesced across lanes) ----
    for (int j = 0; j < 4; ++j) {
        const int nt = wave * 4 + j;
        const int n0 = nt * 16 + l16;
        v8f acc = {};
        for (int k = 0; k < N_C; k += 4) {
            const int kb = k + 2 * half;
            v2f a = { Ss[l16 * SS_STRIDE + kb], Ss[l16 * SS_STRIDE + kb + 1] };
            v2f bv = { V[((size_t)b * N_C + kb) * E_C + n0],
                       V[((size_t)b * N_C + kb + 1) * E_C + n0] };
            acc = wmma4(a, bv, acc);
        }
#pragma unroll
        for (int v = 0; v < 8; ++v)
            CTX[(size_t)(rowbase + v + 8 * half) * E_C + n0] = acc[v];
    }
}

// ---------------------------------------------------------------------------
// Kernel 3: out = LayerNorm(node_fts + CTX @ Wc) * gamma + beta
// ---------------------------------------------------------------------------
__global__ void out_kernel(const float* __restrict__ CTX, const float* __restrict__ Wc,
                           const float* __restrict__ X, const float* __restrict__ gamma,
                           const float* __restrict__ beta, float* __restrict__ out) {
    extern __shared__ float lds[];
    float* Cs  = lds;                       // 16 x XS_STRIDE
    float* Ys  = Cs + 16 * XS_STRIDE;       // 16 x XS_STRIDE
    float* red = Ys + 16 * XS_STRIDE;       // 2 x 256

    const int rowbase = blockIdx.x * 16;
    const int tid = threadIdx.x;

    stage_tile_async(Cs, CTX + (size_t)rowbase * E_C, tid);

    const int lane = tid & 31, wave = tid >> 5;
    const int half = lane >> 4, l16 = lane & 15;

    for (int j = 0; j < 4; ++j) {
        const int nt = wave * 4 + j;
        const int n0 = nt * 16 + l16;
        v8f acc = {};
        for (int k = 0; k < E_C; k += 4) {
            const int kb = k + 2 * half;
            v2f a = { Cs[l16 * XS_STRIDE + kb], Cs[l16 * XS_STRIDE + kb + 1] };
            v2f b = { Wc[(size_t)kb * E_C + n0], Wc[(size_t)(kb + 1) * E_C + n0] };
            acc = wmma4(a, b, acc);
        }
#pragma unroll
        for (int v = 0; v < 8; ++v)
            Ys[(v + 8 * half) * XS_STRIDE + n0] = acc[v];
    }
    __syncthreads();

    // LayerNorm: 16 threads per row, 32 cols each
    const int r = tid >> 4, seg = tid & 15;
    const size_t grow = (size_t)(rowbase + r) * E_C;
    float s1 = 0.0f, s2 = 0.0f;
    for (int c = seg * 32; c < seg * 32 + 32; ++c) {
        const float x = X[grow + c] + Ys[r * XS_STRIDE + c];
        s1 += x; s2 += x * x;
    }
    red[r * 16 + seg] = s1;
    red[256 + r * 16 + seg] = s2;
    __syncthreads();
    float su = 0.0f, sq = 0.0f;
    for (int i = 0; i < 16; ++i) { su += red[r * 16 + i]; sq += red[256 + r * 16 + i]; }
    const float mu   = su * (1.0f / E_C);
    const float var  = sq * (1.0f / E_C) - mu * mu;
    const float rstd = rsqrtf(var + 1e-5f);
    for (int c = seg * 32; c < seg * 32 + 32; ++c) {
        const float x = X[grow + c] + Ys[r * XS_STRIDE + c];
        out[grow + c] = (x - mu) * rstd * gamma[c] + beta[c];
    }
}

extern "C" void kernel_launch(void* const* d_in, const int* in_sizes, int n_in,
                              void* d_out, int out_size, void* d_ws, size_t ws_size,
                              hipStream_t stream) {
    const float* node_fts = (const float*)d_in[0];
    const float* rel      = (const float*)d_in[1];
    const float* Wq       = (const float*)d_in[2];
    const float* bq       = (const float*)d_in[3];
    const float* Wk       = (const float*)d_in[4];
    const float* bk       = (const float*)d_in[5];
    const float* Wc       = (const float*)d_in[6];
    const float* gamma    = (const float*)d_in[7];
    const float* beta     = (const float*)d_in[8];
    float* out = (float*)d_out;

    const size_t BNE = (size_t)B_C * N_C * E_C;   // 8.39M elements
    float* Q   = (float*)d_ws;
    float* K   = Q + BNE;
    float* CTX = K + BNE;                          // total ws: ~100.7 MB

    dim3 grid(B_C * N_C / 16);
    dim3 block(256);

    const size_t lds1 = (size_t)(16 * XS_STRIDE) * sizeof(float);                        // ~33 KB
    const size_t lds2 = (size_t)(16 * XS_STRIDE + 16 * SS_STRIDE + 512) * sizeof(float); // ~164 KB (<320KB WGP LDS)
    const size_t lds3 = (size_t)(2 * 16 * XS_STRIDE + 512) * sizeof(float);              // ~66 KB

    qk_proj_kernel<<<grid, block, lds1, stream>>>(node_fts, Wq, bq, Wk, bk, Q, K);
    attn_kernel<<<grid, block, lds2, stream>>>(Q, K, node_fts, rel, CTX);
    out_kernel<<<grid, block, lds3, stream>>>(CTX, Wc, node_fts, gamma, beta, out);
}